// PR2d_24361054503363
// MI455X (gfx1250) — compile-verified
//
#include <hip/hip_runtime.h>
#include <math.h>

// ---------------------------------------------------------------------------
// CDNA5 (gfx1250) implementation of the rational-FNO layer.
// All contractions are 16x16 (14 zero-padded) complex matmuls executed with
// V_WMMA_F32_16X16X4_F32 (fp32 in/out, K=4), one wave32 per output tile.
// Operands are loaded once into register fragments (64-bit loads where the
// lane's pair is contiguous) and reused across the 4 real products of each
// complex matmul. Wave-private LDS round-trips rely on the ISA's in-order DS
// guarantee instead of block barriers.
// ---------------------------------------------------------------------------

typedef __attribute__((ext_vector_type(2))) float v2f;
typedef __attribute__((ext_vector_type(8))) float v8f;

#define NB   64
#define NCI  64
#define NCO  64
#define NR   14
#define PLANE (4096 * 256)          // one [64*64][16][16] fp32 plane

// workspace plane indices
#define W_ALR  0   // alpha real   [b*64+i]
#define W_ALI  1
#define W_AR   2   // A = 1/(lam - pole1)  [i*64+k], rows o, cols p
#define W_AI   3
#define W_BR   4   // B = 1/(lam - pole2)  rows x, cols q
#define W_BI   5
#define W_E1R  6   // E1 = exp(pole1*t)    rows p, cols z
#define W_E1I  7
#define W_E2R  8   // E2 = exp(pole2*t)    rows q, cols w
#define W_E2I  9
#define W_RR  10   // residue deinterleaved, rows p cols q
#define W_RI  11
#define W_GR  12   // G = A R B^T          rows o, cols x
#define W_GI  13
#define W_R1R 14   // res1 [b*64+k]
#define W_R1I 15
#define W_R2R 16   // res2 [b*64+k]
#define W_R2I 17
#define W_X1  18   // x1 (real)            rows z, cols w
#define F_OFF (19 * PLANE)          // F real (256), then F imag (256)

#define PIF 3.14159265358979323846f

__device__ __forceinline__ v8f vzero() {
  v8f z;
#pragma unroll
  for (int i = 0; i < 8; ++i) z[i] = 0.f;
  return z;
}

// Compiler-level reordering fence. Same-wave DS instructions execute in order
// on CDNA5 (DScnt, in-order), so wave-private LDS store->reload needs no
// workgroup barrier, only a guarantee the compiler keeps program order.
__device__ __forceinline__ void lds_fence() { asm volatile("" ::: "memory"); }

// Register fragments: 4 K-chunks of a 16x16 operand (CDNA5 f32 WMMA layout:
// A chunk = 2 VGPRs, lanes 0-15 hold K=2h / K=2h+1; B chunk row-striped).
struct Frag { v2f c[4]; };

// Left (A) operand. T=0: tile stored row-major [m][k]  -> contiguous v2f load.
//                  T=1: operand is stored-tile^T       -> two strided loads.
template <int T>
__device__ __forceinline__ Frag load_a(const float* X) {
  const int lane = threadIdx.x & 31;
  const int half = lane >> 4;
  const int l    = lane & 15;
  Frag f;
#pragma unroll
  for (int j = 0; j < 4; ++j) {
    const int kx = 4 * j + 2 * half;
    if (T) {
      f.c[j].x = X[kx * 16 + l];
      f.c[j].y = X[(kx + 1) * 16 + l];
    } else {
      f.c[j] = ((const v2f*)X)[l * 8 + 2 * j + half];
    }
  }
  return f;
}

// Right (B) operand. T=0: tile stored row-major [k][n] -> two strided loads.
//                    T=1: operand is stored-tile^T     -> contiguous v2f load.
template <int T>
__device__ __forceinline__ Frag load_b(const float* Y) {
  const int lane = threadIdx.x & 31;
  const int half = lane >> 4;
  const int l    = lane & 15;
  Frag f;
#pragma unroll
  for (int j = 0; j < 4; ++j) {
    const int kx = 4 * j + 2 * half;
    if (T) {
      f.c[j] = ((const v2f*)Y)[l * 8 + 2 * j + half];
    } else {
      f.c[j].x = Y[kx * 16 + l];
      f.c[j].y = Y[(kx + 1) * 16 + l];
    }
  }
  return f;
}

__device__ __forceinline__ v8f wmma_ff(const Frag& a, const Frag& b, v8f acc) {
#pragma unroll
  for (int j = 0; j < 4; ++j)
    acc = __builtin_amdgcn_wmma_f32_16x16x4_f32(false, a.c[j], false, b.c[j],
                                                (short)0, acc, false, false);
  return acc;
}

__device__ __forceinline__ void store_tile(float* P, v8f acc) {
  const int lane = threadIdx.x & 31;
  const int half = lane >> 4;
  const int l    = lane & 15;
#pragma unroll
  for (int v = 0; v < 8; ++v) P[(v + 8 * half) * 16 + l] = acc[v];
}

// ---------------------------------------------------------------------------
// K0: precompute A, B, E1, E2, deinterleave residue, build DFT matrix F.
// ---------------------------------------------------------------------------
__global__ void k_precompute(const float* __restrict__ pole1,
                             const float* __restrict__ pole2,
                             const float* __restrict__ residue,
                             float* __restrict__ ws) {
  const int ik = blockIdx.x;          // i*64 + k
  const int t  = threadIdx.x;         // 0..255
  const int r  = t >> 4, c = t & 15;
  const int tb = ik * 256 + t;
  const float LAMC = 2.f * PIF * 27.f / 14.f;   // 2*pi / (n*d), d = 1/27

  // A: 1/(lam[o=r] - pole1[ik, p=c])
  {
    float ar = 0.f, ai = 0.f;
    if (r < NR) {
      float lam = LAMC * (float)((r <= 6) ? r : r - 14);
      float pr = pole1[(ik * 16 + c) * 2 + 0];
      float pi = pole1[(ik * 16 + c) * 2 + 1];
      float a_ = -pr, b_ = lam - pi;
      float d  = a_ * a_ + b_ * b_;
      ar =  a_ / d;  ai = -b_ / d;
    }
    ws[W_AR * PLANE + tb] = ar;  ws[W_AI * PLANE + tb] = ai;
  }
  // B: 1/(lam[x=r] - pole2[ik, q=c])
  {
    float br = 0.f, bi = 0.f;
    if (r < NR) {
      float lam = LAMC * (float)((r <= 6) ? r : r - 14);
      float pr = pole2[(ik * 16 + c) * 2 + 0];
      float pi = pole2[(ik * 16 + c) * 2 + 1];
      float a_ = -pr, b_ = lam - pi;
      float d  = a_ * a_ + b_ * b_;
      br =  a_ / d;  bi = -b_ / d;
    }
    ws[W_BR * PLANE + tb] = br;  ws[W_BI * PLANE + tb] = bi;
  }
  // E1: exp(pole1[ik, p=r] * t_z), z=c
  {
    float er = 0.f, ei = 0.f;
    if (c < NR) {
      float tz = (float)c / 27.f;
      float pr = pole1[(ik * 16 + r) * 2 + 0];
      float pi = pole1[(ik * 16 + r) * 2 + 1];
      float m  = expf(pr * tz);
      er = m * cosf(pi * tz);  ei = m * sinf(pi * tz);
    }
    ws[W_E1R * PLANE + tb] = er;  ws[W_E1I * PLANE + tb] = ei;
  }
  // E2: exp(pole2[ik, q=r] * t_w), w=c
  {
    float er = 0.f, ei = 0.f;
    if (c < NR) {
      float tw = (float)c / 27.f;
      float pr = pole2[(ik * 16 + r) * 2 + 0];
      float pi = pole2[(ik * 16 + r) * 2 + 1];
      float m  = expf(pr * tw);
      er = m * cosf(pi * tw);  ei = m * sinf(pi * tw);
    }
    ws[W_E2R * PLANE + tb] = er;  ws[W_E2I * PLANE + tb] = ei;
  }
  // residue deinterleave
  ws[W_RR * PLANE + tb] = residue[tb * 2 + 0];
  ws[W_RI * PLANE + tb] = residue[tb * 2 + 1];

  // DFT matrix F[o][t'] = exp(-2*pi*j*o*t'/14), padded with zeros
  if (ik == 0) {
    float fr = 0.f, fi = 0.f;
    if (r < NR && c < NR) {
      float ang = -2.f * PIF * (float)(r * c) / 14.f;
      fr = cosf(ang);  fi = sinf(ang);
    }
    ws[F_OFF + t]       = fr;
    ws[F_OFF + 256 + t] = fi;
  }
}

// ---------------------------------------------------------------------------
// K1: alpha[b,i] = F @ x[b,i] @ F   (F symmetric). One wave per (b,i).
// ---------------------------------------------------------------------------
__global__ void k_alpha(const float* __restrict__ x, float* __restrict__ ws) {
  __shared__ __align__(16) float sh[8][768];   // per wave: X | Tr | Ti
  const int w    = threadIdx.x >> 5;
  const int lane = threadIdx.x & 31;
  const int pair = blockIdx.x * 8 + w;         // b*64 + i
  float* sX  = &sh[w][0];
  float* sTr = &sh[w][256];
  float* sTi = &sh[w][512];

  for (int e = lane; e < 256; e += 32) {
    int r = e >> 4, c = e & 15;
    sX[e] = (r < NR && c < NR) ? x[pair * (NR * NR) + r * NR + c] : 0.f;
  }
  lds_fence();

  const float* FR = ws + F_OFF;
  const float* FI = ws + F_OFF + 256;

  Frag fRa = load_a<0>(FR), fIa = load_a<0>(FI);
  Frag xb  = load_b<0>(sX);
  v8f tr = wmma_ff(fRa, xb, vzero());
  v8f ti = wmma_ff(fIa, xb, vzero());
  store_tile(sTr, tr);
  store_tile(sTi, ti);
  lds_fence();

  Frag trf = load_a<0>(sTr), tif = load_a<0>(sTi);
  Frag fRb = load_b<0>(FR), fIb = load_b<0>(FI);
  v8f p1 = wmma_ff(trf, fRb, vzero());
  v8f p2 = wmma_ff(tif, fIb, vzero());
  v8f ai = wmma_ff(trf, fIb, vzero());
  ai     = wmma_ff(tif, fRb, ai);
  v8f ar;
#pragma unroll
  for (int v = 0; v < 8; ++v) ar[v] = p1[v] - p2[v];

  store_tile(ws + W_ALR * PLANE + pair * 256, ar);
  store_tile(ws + W_ALI * PLANE + pair * 256, ai);
}

// ---------------------------------------------------------------------------
// K2: res2[b,k] = sum_i residue_ik (.) (A_ik^T @ alpha_bi @ B_ik)
// One wave per (b,k), loop over i. Heaviest stage (~8.4M WMMA total).
// ---------------------------------------------------------------------------
__global__ void k_res2(float* __restrict__ ws) {
  __shared__ __align__(16) float sh[8][512];   // per wave: Ur | Ui
  const int w    = threadIdx.x >> 5;
  const int lane = threadIdx.x & 31;
  const int half = lane >> 4, l = lane & 15;
  const int pair = blockIdx.x * 8 + w;         // b*64 + k
  const int b = pair >> 6, k = pair & 63;
  float* sUr = &sh[w][0];
  float* sUi = &sh[w][256];

  v8f accR = vzero(), accI = vzero();
  for (int i = 0; i < NCI; ++i) {
    const int ik = i * 64 + k;
    const float* aR  = ws + W_ALR * PLANE + (b * 64 + i) * 256;
    const float* aI  = ws + W_ALI * PLANE + (b * 64 + i) * 256;
    const float* BR_ = ws + W_BR * PLANE + ik * 256;
    const float* BI_ = ws + W_BI * PLANE + ik * 256;
    const float* AR_ = ws + W_AR * PLANE + ik * 256;
    const float* AI_ = ws + W_AI * PLANE + ik * 256;
    const float* RR_ = ws + W_RR * PLANE + ik * 256;
    const float* RI_ = ws + W_RI * PLANE + ik * 256;

    // U = alpha @ B  (complex), fragments loaded once
    Frag aRf = load_a<0>(aR),  aIf = load_a<0>(aI);
    Frag bRf = load_b<0>(BR_), bIf = load_b<0>(BI_);
    v8f u1 = wmma_ff(aRf, bRf, vzero());
    v8f u2 = wmma_ff(aIf, bIf, vzero());
    v8f ui = wmma_ff(aRf, bIf, vzero());
    ui     = wmma_ff(aIf, bRf, ui);
    v8f ur;
#pragma unroll
    for (int v = 0; v < 8; ++v) ur[v] = u1[v] - u2[v];
    store_tile(sUr, ur);
    store_tile(sUi, ui);
    lds_fence();

    // M = A^T @ U  (complex)
    Frag pRf = load_a<1>(AR_), pIf = load_a<1>(AI_);
    Frag uRf = load_b<0>(sUr), uIf = load_b<0>(sUi);
    v8f m1 = wmma_ff(pRf, uRf, vzero());
    v8f m2 = wmma_ff(pIf, uIf, vzero());
    v8f mi = wmma_ff(pRf, uIf, vzero());
    mi     = wmma_ff(pIf, uRf, mi);

    // acc += residue (.) M   (elementwise complex)
#pragma unroll
    for (int v = 0; v < 8; ++v) {
      int p = v + 8 * half, q = l;
      float rr = RR_[p * 16 + q], ri = RI_[p * 16 + q];
      float mr = m1[v] - m2[v];
      accR[v] += rr * mr - ri * mi[v];
      accI[v] += rr * mi[v] + ri * mr;
    }
    lds_fence();
  }
  store_tile(ws + W_R2R * PLANE + pair * 256, accR);
  store_tile(ws + W_R2I * PLANE + pair * 256, accI);
}

// ---------------------------------------------------------------------------
// K3a: G_ik = A_ik @ R_ik @ B_ik^T  (complex). One wave per (i,k).
// ---------------------------------------------------------------------------
__global__ void k_G(float* __restrict__ ws) {
  __shared__ __align__(16) float sh[8][512];
  const int w  = threadIdx.x >> 5;
  const int ik = blockIdx.x * 8 + w;
  float* sWr = &sh[w][0];
  float* sWi = &sh[w][256];
  const float* AR_ = ws + W_AR * PLANE + ik * 256;
  const float* AI_ = ws + W_AI * PLANE + ik * 256;
  const float* BR_ = ws + W_BR * PLANE + ik * 256;
  const float* BI_ = ws + W_BI * PLANE + ik * 256;
  const float* RR_ = ws + W_RR * PLANE + ik * 256;
  const float* RI_ = ws + W_RI * PLANE + ik * 256;

  Frag aRf = load_a<0>(AR_), aIf = load_a<0>(AI_);
  Frag rRf = load_b<0>(RR_), rIf = load_b<0>(RI_);
  v8f w1 = wmma_ff(aRf, rRf, vzero());
  v8f w2 = wmma_ff(aIf, rIf, vzero());
  v8f wi = wmma_ff(aRf, rIf, vzero());
  wi     = wmma_ff(aIf, rRf, wi);
  v8f wr;
#pragma unroll
  for (int v = 0; v < 8; ++v) wr[v] = w1[v] - w2[v];
  store_tile(sWr, wr);
  store_tile(sWi, wi);
  lds_fence();

  Frag wRf = load_a<0>(sWr), wIf = load_a<0>(sWi);
  Frag bRt = load_b<1>(BR_), bIt = load_b<1>(BI_);   // B^T, contiguous loads
  v8f g1 = wmma_ff(wRf, bRt, vzero());
  v8f g2 = wmma_ff(wIf, bIt, vzero());
  v8f gi = wmma_ff(wRf, bIt, vzero());
  gi     = wmma_ff(wIf, bRt, gi);
  v8f gr;
#pragma unroll
  for (int v = 0; v < 8; ++v) gr[v] = g1[v] - g2[v];
  store_tile(ws + W_GR * PLANE + ik * 256, gr);
  store_tile(ws + W_GI * PLANE + ik * 256, gi);
}

// ---------------------------------------------------------------------------
// K3b: res1[b,k,o,x] = sum_i alpha[b,i,o,x] * G[i,k,o,x]  (elementwise VALU)
// ---------------------------------------------------------------------------
__global__ void k_res1(float* __restrict__ ws) {
  const int bk = blockIdx.x;
  const int t  = threadIdx.x;
  const int b = bk >> 6, k = bk & 63;
  const int r = t >> 4, c = t & 15;
  float sr = 0.f, si = 0.f;
  if (r < NR && c < NR) {
    for (int i = 0; i < NCI; ++i) {
      float ar = ws[W_ALR * PLANE + (b * 64 + i) * 256 + t];
      float ai = ws[W_ALI * PLANE + (b * 64 + i) * 256 + t];
      float gr = ws[W_GR * PLANE + (i * 64 + k) * 256 + t];
      float gi = ws[W_GI * PLANE + (i * 64 + k) * 256 + t];
      sr += ar * gr - ai * gi;
      si += ar * gi + ai * gr;
    }
  }
  ws[W_R1R * PLANE + bk * 256 + t] = sr;
  ws[W_R1I * PLANE + bk * 256 + t] = si;
}

// ---------------------------------------------------------------------------
// K3c: x1[b,k] = Re( conj(F) @ res1[b,k] @ conj(F) ) / 196
// ---------------------------------------------------------------------------
__global__ void k_x1(float* __restrict__ ws) {
  __shared__ __align__(16) float sh[8][512];
  const int w  = threadIdx.x >> 5;
  const int bk = blockIdx.x * 8 + w;
  float* sWr = &sh[w][0];
  float* sWi = &sh[w][256];
  const float* FR  = ws + F_OFF;
  const float* FI  = ws + F_OFF + 256;
  const float* R1R = ws + W_R1R * PLANE + bk * 256;
  const float* R1I = ws + W_R1I * PLANE + bk * 256;

  // W = conj(F) @ R : real = Fr Rr + Fi Ri ; imag = Fr Ri - Fi Rr
  Frag fRa = load_a<0>(FR), fIa = load_a<0>(FI);
  Frag rRf = load_b<0>(R1R), rIf = load_b<0>(R1I);
  v8f wr = wmma_ff(fRa, rRf, vzero());
  wr     = wmma_ff(fIa, rIf, wr);
  v8f i1 = wmma_ff(fRa, rIf, vzero());
  v8f i2 = wmma_ff(fIa, rRf, vzero());
  v8f wi;
#pragma unroll
  for (int v = 0; v < 8; ++v) wi[v] = i1[v] - i2[v];
  store_tile(sWr, wr);
  store_tile(sWi, wi);
  lds_fence();

  // x1 = Re(W @ conj(F)) = Wr Fr + Wi Fi   (F symmetric)
  Frag wRf = load_a<0>(sWr), wIf = load_a<0>(sWi);
  Frag fRb = load_b<0>(FR), fIb = load_b<0>(FI);
  v8f xx = wmma_ff(wRf, fRb, vzero());
  xx     = wmma_ff(wIf, fIb, xx);
#pragma unroll
  for (int v = 0; v < 8; ++v) xx[v] *= (1.f / 196.f);
  store_tile(ws + W_X1 * PLANE + bk * 256, xx);
}

// ---------------------------------------------------------------------------
// K4: out[b,c] = x1[b,c] + Re( sum_j E1_jc^T @ res2[b,j] @ E2_jc ) / 196
// One wave per (b,c), loop over j. (~6.3M WMMA total)
// ---------------------------------------------------------------------------
__global__ void k_x2(const float* __restrict__ ws_c, float* __restrict__ ws,
                     float* __restrict__ out) {
  __shared__ __align__(16) float sh[8][512];
  const int w    = threadIdx.x >> 5;
  const int lane = threadIdx.x & 31;
  const int half = lane >> 4, l = lane & 15;
  const int pair = blockIdx.x * 8 + w;       // b*64 + c
  const int b = pair >> 6, c = pair & 63;
  float* sWr = &sh[w][0];
  float* sWi = &sh[w][256];

  v8f accP = vzero(), accM = vzero();
  for (int j = 0; j < NCO; ++j) {
    const int jc = j * 64 + c;
    const float* E1R_ = ws_c + W_E1R * PLANE + jc * 256;
    const float* E1I_ = ws_c + W_E1I * PLANE + jc * 256;
    const float* E2R_ = ws_c + W_E2R * PLANE + jc * 256;
    const float* E2I_ = ws_c + W_E2I * PLANE + jc * 256;
    const float* R2R_ = ws_c + W_R2R * PLANE + (b * 64 + j) * 256;
    const float* R2I_ = ws_c + W_R2I * PLANE + (b * 64 + j) * 256;

    // W = E1^T @ res2   (complex)
    Frag e1Rf = load_a<1>(E1R_), e1If = load_a<1>(E1I_);
    Frag r2Rf = load_b<0>(R2R_), r2If = load_b<0>(R2I_);
    v8f w1 = wmma_ff(e1Rf, r2Rf, vzero());
    v8f w2 = wmma_ff(e1If, r2If, vzero());
    v8f wi = wmma_ff(e1Rf, r2If, vzero());
    wi     = wmma_ff(e1If, r2Rf, wi);
    v8f wr;
#pragma unroll
    for (int v = 0; v < 8; ++v) wr[v] = w1[v] - w2[v];
    store_tile(sWr, wr);
    store_tile(sWi, wi);
    lds_fence();

    // Re(W @ E2): accP += Wr E2r, accM += Wi E2i
    Frag wRf = load_a<0>(sWr), wIf = load_a<0>(sWi);
    Frag e2Rf = load_b<0>(E2R_), e2If = load_b<0>(E2I_);
    accP = wmma_ff(wRf, e2Rf, accP);
    accM = wmma_ff(wIf, e2If, accM);
    lds_fence();
  }

#pragma unroll
  for (int v = 0; v < 8; ++v) {
    int z = v + 8 * half, ww = l;
    if (z < NR && ww < NR) {
      float x1v = ws[W_X1 * PLANE + pair * 256 + z * 16 + ww];
      out[(pair * NR + z) * NR + ww] = x1v + (accP[v] - accM[v]) * (1.f / 196.f);
    }
  }
}

// ---------------------------------------------------------------------------
extern "C" void kernel_launch(void* const* d_in, const int* in_sizes, int n_in,
                              void* d_out, int out_size, void* d_ws, size_t ws_size,
                              hipStream_t stream) {
  const float* x     = (const float*)d_in[0];   // [64,64,14,14] f32
  const float* pole1 = (const float*)d_in[1];   // [64,64,16]    c64 (interleaved)
  const float* pole2 = (const float*)d_in[2];   // [64,64,16]    c64
  const float* resid = (const float*)d_in[3];   // [64,64,16,16] c64
  float* out = (float*)d_out;                   // [64,64,14,14] f32
  float* ws  = (float*)d_ws;                    // needs ~80 MB

  k_precompute<<<4096, 256, 0, stream>>>(pole1, pole2, resid, ws);
  k_alpha     <<< 512, 256, 0, stream>>>(x, ws);
  k_res2      <<< 512, 256, 0, stream>>>(ws);
  k_G         <<< 512, 256, 0, stream>>>(ws);
  k_res1      <<<4096, 256, 0, stream>>>(ws);
  k_x1        <<< 512, 256, 0, stream>>>(ws);
  k_x2        <<< 512, 256, 0, stream>>>(ws, ws, out);
}